// RadiusSearch_75660143886577
// MI455X (gfx1250) — compile-verified
//
#include <hip/hip_runtime.h>

// RadiusSearch on MI455X (gfx1250, wave32):
//   pass 0: init padding (-1 for indices, 0 for distances), zero counts
//   pass 1: per-query neighbor counts via V_WMMA_F32_16X16X4_F32 distance tiles
//   pass 2: exclusive scan of counts -> row_splits (int in ws, float in d_out)
//   pass 3: ordered compaction via identical WMMA tiles + ballot/popcount
//
// d_out (float) layout: [0,CAP) nbr_index | [CAP, CAP+Q+1) row_splits | [.., +CAP) nbr_dist

#define NPTS 32768
#define NQRY 8192
#define CAP  (NQRY * 128)
#define NTILES (NPTS / 16)   // 2048 point tiles of 16

typedef __attribute__((ext_vector_type(2))) float v2f;
typedef __attribute__((ext_vector_type(8))) float v8f;

// ---------------------------------------------------------------- init ------
__global__ void rs_fill_kernel(float* __restrict__ out, int* __restrict__ counts) {
  long i = (long)blockIdx.x * blockDim.x + threadIdx.x;
  long stride = (long)gridDim.x * blockDim.x;
  for (long k = i; k < CAP; k += stride) {
    out[k] = -1.0f;                      // nbr_index padding
    out[CAP + NQRY + 1 + k] = 0.0f;      // nbr_dist padding
  }
  for (long k = i; k < NQRY; k += stride) counts[k] = 0;
}

// ------------------------------------------------- shared tile-prep helpers -
// A-tile (16 queries x K=4, K=3 zero-padded) in ISA 32-bit A layout:
//   lanes 0-15: K=0,1 of row M=lane ; lanes 16-31: K=2,3(pad) of row M=lane-16
__device__ __forceinline__ v2f load_a_tile(const float* Qr, int qbase, int lm, int half) {
  const float* qrow = Qr + 3 * (qbase + lm);
  v2f a;
  a.x = qrow[2 * half];                  // K=0 (half 0) or K=2 (half 1)
  a.y = half ? 0.0f : qrow[1];           // K=1 or zero pad (K=3)
  return a;
}

// q^2 and r^2 arranged in the C layout: slot v <-> row M = v + 8*half
__device__ __forceinline__ void load_row_vectors(const float* Qr, const float* radii,
                                                 int qbase, int half,
                                                 float q2c[8], float r2c[8]) {
#pragma unroll
  for (int v = 0; v < 8; ++v) {
    int row = qbase + v + 8 * half;
    const float* qr = Qr + 3 * row;
    q2c[v] = qr[0] * qr[0] + qr[1] * qr[1] + qr[2] * qr[2];
    float r = radii[row];
    r2c[v] = r * r;
  }
}

// ---------------------------------------------------------------- pass 1 ----
// grid (NQRY/16, 8), block 32 (one wave). Each wave: 16-query strip x 4096 pts.
__global__ void rs_count_kernel(const float* __restrict__ P,
                                const float* __restrict__ Qr,
                                const float* __restrict__ radii,
                                int* __restrict__ counts,
                                int tilesPerChunk) {
  const int lane = threadIdx.x;
  const int half = lane >> 4;
  const int lm   = lane & 15;
  const int qbase = blockIdx.x * 16;
  const int nbase0 = blockIdx.y * tilesPerChunk * 16;

  v2f a = load_a_tile(Qr, qbase, lm, half);
  float q2c[8], r2c[8];
  load_row_vectors(Qr, radii, qbase, half, q2c, r2c);

  int cnt[8];
#pragma unroll
  for (int v = 0; v < 8; ++v) cnt[v] = 0;

  for (int t = 0; t < tilesPerChunk; ++t) {
    const int col = nbase0 + t * 16 + lm;           // C-layout column for this lane
    const float* pr = P + 3 * col;
    float px = pr[0], py = pr[1], pz = pr[2];
    float p2 = px * px + py * py + pz * pz;
    v2f b;                                          // B 4x16: K-halves split by lane half
    b.x = half ? pz : px;                           // K=0 | K=2
    b.y = half ? 0.0f : py;                         // K=1 | K=3 (pad)
    v8f c = {};
    c = __builtin_amdgcn_wmma_f32_16x16x4_f32(false, a, false, b,
                                              (short)0, c, false, false);
#pragma unroll
    for (int v = 0; v < 8; ++v) {
      float d2 = fmaf(-2.0f, c[v], q2c[v] + p2);    // |q|^2 + |p|^2 - 2 q.p
      cnt[v] += (d2 <= r2c[v]) ? 1 : 0;
    }
  }

  // reduce columns within each 16-lane half
#pragma unroll
  for (int v = 0; v < 8; ++v) {
#pragma unroll
    for (int m = 1; m < 16; m <<= 1) cnt[v] += __shfl_xor(cnt[v], m, 32);
  }
  if (lm == 0) {                                    // lanes 0 and 16
#pragma unroll
    for (int v = 0; v < 8; ++v)
      atomicAdd(&counts[qbase + v + 8 * half], cnt[v]);
  }
}

// ---------------------------------------------------------------- pass 2 ----
// single block, 256 threads, 32 elements each: exclusive scan of counts[8192]
__global__ void rs_scan_kernel(const int* __restrict__ counts,
                               int* __restrict__ rs_int,
                               float* __restrict__ rs_float) {
  __shared__ int part[256];
  const int t = threadIdx.x;
  const int base = t * 32;
  int local[32];
  int s = 0;
#pragma unroll
  for (int i = 0; i < 32; ++i) { local[i] = s; s += counts[base + i]; }
  part[t] = s;
  __syncthreads();
  for (int off = 1; off < 256; off <<= 1) {         // Hillis-Steele inclusive scan
    int v = part[t];
    int add = (t >= off) ? part[t - off] : 0;
    __syncthreads();
    part[t] = v + add;
    __syncthreads();
  }
  int offset = (t == 0) ? 0 : part[t - 1];
#pragma unroll
  for (int i = 0; i < 32; ++i) {
    int val = offset + local[i];
    rs_int[base + i] = val;
    rs_float[base + i] = (float)val;
  }
  if (t == 255) { rs_int[NQRY] = part[255]; rs_float[NQRY] = (float)part[255]; }
}

// ---------------------------------------------------------------- pass 3 ----
// grid NQRY/16, block 32. Same WMMA tiles as pass 1 (bitwise-identical preds);
// LDS bounce remaps row-in-VGPR -> row-per-lane-half for ballot compaction.
__global__ void rs_emit_kernel(const float* __restrict__ P,
                               const float* __restrict__ Qr,
                               const float* __restrict__ radii,
                               const int* __restrict__ rs,
                               float* __restrict__ out_idx,
                               float* __restrict__ out_dist) {
  __shared__ float tile[256];
  const int lane = threadIdx.x;
  const int half = lane >> 4;
  const int lm   = lane & 15;
  const int qbase = blockIdx.x * 16;

  v2f a = load_a_tile(Qr, qbase, lm, half);
  float q2c[8], r2c[8];
  load_row_vectors(Qr, radii, qbase, half, q2c, r2c);

  int off[8], end[8];                               // running slot per owned row
#pragma unroll
  for (int j = 0; j < 8; ++j) {
    int row = qbase + j + 8 * half;
    off[j] = rs[row];
    end[j] = rs[row + 1];
  }
  const unsigned lt_mask = (1u << lm) - 1u;

  for (int t = 0; t < NTILES; ++t) {
    const int nbase = t * 16;
    const int col = nbase + lm;
    const float* pr = P + 3 * col;
    float px = pr[0], py = pr[1], pz = pr[2];
    float p2 = px * px + py * py + pz * pz;
    v2f b;
    b.x = half ? pz : px;
    b.y = half ? 0.0f : py;
    v8f c = {};
    c = __builtin_amdgcn_wmma_f32_16x16x4_f32(false, a, false, b,
                                              (short)0, c, false, false);
#pragma unroll
    for (int v = 0; v < 8; ++v) {
      float d2 = fmaf(-2.0f, c[v], q2c[v] + p2);
      tile[(v + 8 * half) * 16 + lm] = d2;          // row-major 16x16 in LDS
    }
    __syncthreads();
#pragma unroll
    for (int j = 0; j < 8; ++j) {                   // lanes 0-15: row j, 16-31: row j+8
      float d = tile[(j + 8 * half) * 16 + lm];
      bool pred = d <= r2c[j];
      unsigned bal = __builtin_amdgcn_ballot_w32(pred);
      unsigned my = half ? (bal >> 16) : (bal & 0xffffu);
      int pos = off[j] + __builtin_popcount(my & lt_mask);
      if (pred && pos < end[j]) {
        out_idx[pos]  = (float)col;                 // point index, exact in fp32
        out_dist[pos] = fmaxf(d, 0.0f) / r2c[j];    // normalized squared distance
      }
      off[j] += __builtin_popcount(my);
    }
    __syncthreads();
  }
}

// ---------------------------------------------------------------- launch ----
extern "C" void kernel_launch(void* const* d_in, const int* in_sizes, int n_in,
                              void* d_out, int out_size, void* d_ws, size_t ws_size,
                              hipStream_t stream) {
  const float* P     = (const float*)d_in[0];   // [32768,3]
  const float* Qr    = (const float*)d_in[1];   // [8192,3]
  const float* radii = (const float*)d_in[2];   // [8192]

  float* out      = (float*)d_out;
  float* out_idx  = out;
  float* rs_f     = out + CAP;
  float* out_dist = out + CAP + NQRY + 1;

  int* counts = (int*)d_ws;                     // [8192]
  int* rs_i   = counts + NQRY;                  // [8193]

  hipLaunchKernelGGL(rs_fill_kernel, dim3(2048), dim3(256), 0, stream, out, counts);
  hipLaunchKernelGGL(rs_count_kernel, dim3(NQRY / 16, 8), dim3(32), 0, stream,
                     P, Qr, radii, counts, NTILES / 8);
  hipLaunchKernelGGL(rs_scan_kernel, dim3(1), dim3(256), 0, stream, counts, rs_i, rs_f);
  hipLaunchKernelGGL(rs_emit_kernel, dim3(NQRY / 16), dim3(32), 0, stream,
                     P, Qr, radii, rs_i, out_idx, out_dist);
}